// Base_75265006895876
// MI455X (gfx1250) — compile-verified
//
#include <hip/hip_runtime.h>
#include <math.h>

// MI455X / gfx1250, wave32. Memory-bound gather workload; WMMA f32 16x16x4
// computes the per-(t,b) 16x16 Gram matrix exactly (A x A^T, same operand
// for A and B thanks to the symmetric lane layout).

typedef __attribute__((ext_vector_type(2))) float v2f;
typedef __attribute__((ext_vector_type(8))) float v8f;

#define T_DIM 17
#define B_DIM 128
#define K_DIM 16
#define L_DIM 24
#define D_DIM 1024
#define LARGE_F 1.0e6f
#define EPS_F 1e-8f

// ---------------------------------------------------------------------------
// Kernel 1: sent[b,:] = sum_{l < cap_len[b]} table[caption[b,l],:]
//           sent_norm[b] = ||sent[b,:]||
// One block per b; 256 threads, each owns 4 consecutive d's (float4).
// ---------------------------------------------------------------------------
__global__ __launch_bounds__(256) void sent_kernel(
    const int* __restrict__ caption,    // [B,L]
    const int* __restrict__ cap_len,    // [B]
    const float* __restrict__ table,    // [V,D]
    float* __restrict__ sent,           // [B,D]  (workspace)
    float* __restrict__ sent_norm)      // [B]    (workspace)
{
    __shared__ float red[256];
    const int b   = blockIdx.x;
    const int tid = threadIdx.x;
    const int len = cap_len[b];
    const int d0  = tid * 4;

    float4 acc = make_float4(0.f, 0.f, 0.f, 0.f);
    for (int l = 0; l < len; ++l) {
        const int c = caption[b * L_DIM + l];
        const float4 r = *(const float4*)(table + (size_t)c * D_DIM + d0);
        acc.x += r.x; acc.y += r.y; acc.z += r.z; acc.w += r.w;
    }
    *(float4*)(sent + (size_t)b * D_DIM + d0) = acc;

    red[tid] = acc.x*acc.x + acc.y*acc.y + acc.z*acc.z + acc.w*acc.w;
    __syncthreads();
    for (int s = 128; s > 0; s >>= 1) {
        if (tid < s) red[tid] += red[tid + s];
        __syncthreads();
    }
    if (tid == 0) sent_norm[b] = sqrtf(red[0]);
}

// ---------------------------------------------------------------------------
// Kernel 2: one wave32 per (t,b) tile.
//   gram = emb(16xD) x emb^T via 256 x V_WMMA_F32_16X16X4_F32
//   sq   = diag(gram); dots[k] = <emb[k,:], sent[b,:]> accumulated in VALU.
// A-matrix 16x4 f32 layout (ISA 7.12.2): lanes 0-15 hold rows 0-15 at
// K=k0,k0+1 (2 VGPRs); lanes 16-31 hold the same rows at K=k0+2,k0+3.
// B = A^T for a Gram matrix has the identical lane layout, so the same
// v2f operand is passed as both A and B.
// ---------------------------------------------------------------------------
__global__ __launch_bounds__(32) void tile_kernel(
    const int* __restrict__ topk,        // [T,B,K]
    const float* __restrict__ table,     // [V,D]
    const float* __restrict__ sent,      // [B,D]
    const float* __restrict__ sent_norm, // [B]
    float* __restrict__ out_min,         // [B,T,K]
    float* __restrict__ out_cos,         // [B,T,K]
    float* __restrict__ out_sc)          // [B,T,K]
{
    __shared__ float sSent[D_DIM];            // 4 KB
    __shared__ float sGram[K_DIM * K_DIM];    // 1 KB
    __shared__ int   sIdx[K_DIM];
    __shared__ float sDot[32];

    const int tile = blockIdx.x;              // t*B + b
    const int t    = tile / B_DIM;
    const int b    = tile - t * B_DIM;
    const int tid  = threadIdx.x;

    if (tid < K_DIM)
        sIdx[tid] = topk[(size_t)t * B_DIM * K_DIM + (size_t)b * K_DIM + tid];

    // Stage sent[b,:] in LDS (32 floats per lane via float4).
    {
        const float4* src = (const float4*)(sent + (size_t)b * D_DIM);
        float4* dst = (float4*)sSent;
        for (int i = tid; i < D_DIM / 4; i += 32)
            dst[i] = src[i];
    }
    __syncthreads();

    const int rowk = tid & 15;                // matrix row this lane feeds
    const int koff = (tid >> 4) * 2;          // K sub-offset: 0 or 2
    const float* rowptr = table + (size_t)sIdx[rowk] * D_DIM + koff;
    const float* sptr   = sSent + koff;

    v8f  acc  = {};
    float dotp = 0.f;
#pragma unroll 4
    for (int kk = 0; kk < D_DIM; kk += 4) {
        const v2f e = *(const v2f*)(rowptr + kk);   // global_load_b64 (gather)
        const v2f s = *(const v2f*)(sptr + kk);     // ds_load_b64 (broadcast)
        dotp += e.x * s.x + e.y * s.y;
        // D = A x B + C, A = B = emb tile slice (Gram).
        acc = __builtin_amdgcn_wmma_f32_16x16x4_f32(
            /*neg_a=*/false, e, /*neg_b=*/false, e,
            /*c_mod=*/(short)0, acc, /*reuse_a=*/false, /*reuse_b=*/false);
    }

    sDot[tid] = dotp;
    // Scatter accumulator to LDS per C/D layout: VGPR r, lane L ->
    //   M = (L<16 ? r : r+8), N = L&15.
    {
        const int M0 = (tid < 16) ? 0 : 8;
        const int N  = tid & 15;
#pragma unroll
        for (int r = 0; r < 8; ++r)
            sGram[(M0 + r) * K_DIM + N] = acc[r];
    }
    __syncthreads();

    if (tid < K_DIM) {
        const int k = tid;
        const float dk  = sDot[k] + sDot[k + 16];      // full <emb[k],sent>
        const float sqk = sGram[k * K_DIM + k];        // ||emb[k]||^2
        const float nk  = sqrtf(sqk);

        // min over i of sqrt(max(sq_i + sq_k - 2*gram[i][k], 1e-12)),
        // diagonal offset by LARGE (matches reference's +LARGE*eye then min).
        float mn = 3.4e38f;
#pragma unroll
        for (int i = 0; i < K_DIM; ++i) {
            const float sqi = sGram[i * K_DIM + i];
            const float g   = sGram[i * K_DIM + k];
            float d2 = fmaxf(sqi + sqk - 2.f * g, 1e-12f);
            float d  = sqrtf(d2);
            if (i == k) d += LARGE_F;
            mn = fminf(mn, d);
        }

        // cosine to element 0
        const float n0 = sqrtf(sGram[0]);
        float cosd = sGram[k] / (fmaxf(n0, EPS_F) * fmaxf(nk, EPS_F));
        if (k == 0) cosd = 0.f;

        // cosine to sentence embedding
        const float sn = sent_norm[b];
        const float sc = dk / (fmaxf(sn, EPS_F) * fmaxf(nk, EPS_F));

        const size_t o = (size_t)b * T_DIM * K_DIM + (size_t)t * K_DIM + k;
        out_min[o] = mn;
        out_cos[o] = cosd;
        out_sc[o]  = sc;
    }
}

// ---------------------------------------------------------------------------
extern "C" void kernel_launch(void* const* d_in, const int* in_sizes, int n_in,
                              void* d_out, int out_size, void* d_ws, size_t ws_size,
                              hipStream_t stream) {
    (void)in_sizes; (void)n_in; (void)out_size; (void)ws_size;

    const int*   topk    = (const int*)d_in[0];    // [T,B,K]
    const int*   caption = (const int*)d_in[1];    // [B,L]
    const int*   cap_len = (const int*)d_in[2];    // [B]
    const float* table   = (const float*)d_in[3];  // [V,D]

    float* sent      = (float*)d_ws;                      // B*D floats
    float* sent_norm = sent + (size_t)B_DIM * D_DIM;      // B floats

    float* out     = (float*)d_out;                       // [3,B,T,K] concat
    float* out_min = out;
    float* out_cos = out + (size_t)B_DIM * T_DIM * K_DIM;
    float* out_sc  = out + 2 * (size_t)B_DIM * T_DIM * K_DIM;

    sent_kernel<<<B_DIM, 256, 0, stream>>>(caption, cap_len, table, sent, sent_norm);
    tile_kernel<<<T_DIM * B_DIM, 32, 0, stream>>>(topk, table, sent, sent_norm,
                                                  out_min, out_cos, out_sc);
}